// MatrixFunctionBlock_3547642987271
// MI455X (gfx1250) — compile-verified
//
#include <hip/hip_runtime.h>
#include <hip/hip_bf16.h>
#include <math.h>

// Problem constants from the reference
#define BB 32
#define FF 64
#define NN 128
#define EPS 1e-9f

typedef __attribute__((ext_vector_type(2))) float v2f;
typedef __attribute__((ext_vector_type(8))) float v8f;

// ---------------------------------------------------------------------------
// Kernel 1: per-(b,f) masked trace(x) and masked trace(x^2).
// Each block handles one 128x128 matrix; wave p (0..7) computes diagonal
// 16x16 block p of x*x with V_WMMA_F32_16X16X4_F32, accumulating over K=128.
// Only the masked diagonal of that block is kept.
// ---------------------------------------------------------------------------
__global__ void __launch_bounds__(256)
mfb_stats_kernel(const float* __restrict__ x, const float* __restrict__ mask,
                 float* __restrict__ trace_out, float* __restrict__ tracesq_out) {
    const int bf = blockIdx.x;              // b*FF + f
    const int b  = bf / FF;
    const int lane = threadIdx.x & 31;
    const int w    = threadIdx.x >> 5;      // wave id = diagonal block p
    const int p    = w;

    const float* __restrict__ X = x + (size_t)bf * NN * NN;

    const int m  = lane & 15;               // M (for A) == N (for B) index
    const int kk = (lane >> 4) * 2;         // K pair selected by lane half

    const float* __restrict__ arow = X + (size_t)(p * 16 + m) * NN; // A: row of x
    const int bcol = p * 16 + m;                                     // B: col of x

    v8f c0 = {}; v8f c1 = {};
    #pragma unroll 4
    for (int k0 = 0; k0 < NN; k0 += 8) {
        // ---- step k0 ----
        v2f a0 = *(const v2f*)(arow + k0 + kk);          // A[m][k0+kk], A[m][k0+kk+1]
        v2f b0;
        b0.x = X[(size_t)(k0 + kk)     * NN + bcol];     // B[kk][n]   = x[k][col]
        b0.y = X[(size_t)(k0 + kk + 1) * NN + bcol];
        c0 = __builtin_amdgcn_wmma_f32_16x16x4_f32(false, a0, false, b0,
                                                   (short)0, c0, false, false);
        // ---- step k0+4 (independent accumulator chain) ----
        v2f a1 = *(const v2f*)(arow + k0 + 4 + kk);
        v2f b1;
        b1.x = X[(size_t)(k0 + 4 + kk)     * NN + bcol];
        b1.y = X[(size_t)(k0 + 4 + kk + 1) * NN + bcol];
        c1 = __builtin_amdgcn_wmma_f32_16x16x4_f32(false, a1, false, b1,
                                                   (short)0, c1, false, false);
    }
    v8f c = c0 + c1;

    // C layout (16x16 f32): VGPR r, lanes 0-15 -> M=r, N=lane;
    //                       lanes 16-31 -> M=r+8, N=lane-16.
    // Diagonal element d lives at: d<8  -> vgpr d,   lane d
    //                              d>=8 -> vgpr d-8, lane d+16
    const bool active = (lane < 8) || (lane >= 24);
    const int  want_r = (lane < 8) ? lane : (lane - 24);
    float diag = 0.f;
    #pragma unroll
    for (int r = 0; r < 8; ++r)
        if (active && want_r == r) diag = c[r];

    float tsq_part = 0.f;
    if (active) {
        const int d = (lane < 8) ? lane : (lane - 16);   // local diag row
        const int i = p * 16 + d;
        const float mi = mask[b * NN + i];
        tsq_part = diag * mi * mi;
    }

    // plain masked trace of x (diag of x itself)
    float tr_part = 0.f;
    if (lane < 16) {
        const int i = p * 16 + lane;
        const float mi = mask[b * NN + i];
        tr_part = X[(size_t)i * NN + i] * mi * mi;
    }

    // wave32 reductions
    #pragma unroll
    for (int off = 16; off > 0; off >>= 1) {
        tsq_part += __shfl_xor(tsq_part, off, 32);
        tr_part  += __shfl_xor(tr_part,  off, 32);
    }

    __shared__ float sTr[8];
    __shared__ float sTs[8];
    if (lane == 0) { sTr[w] = tr_part; sTs[w] = tsq_part; }
    __syncthreads();
    if (threadIdx.x == 0) {
        float tr = 0.f, ts = 0.f;
        #pragma unroll
        for (int q = 0; q < 8; ++q) { tr += sTr[q]; ts += sTs[q]; }
        trace_out[bf]   = tr;
        tracesq_out[bf] = ts;
    }
}

// ---------------------------------------------------------------------------
// Kernel 2: batch statistics -> running stats (momentum warmup) -> fold the
// affine gain into two per-feature scalars: rm[f] and scale[f].
// Single block of 64 threads (t == f).
// ---------------------------------------------------------------------------
__global__ void __launch_bounds__(64)
mfb_finalize_kernel(const float* __restrict__ mask,
                    const float* __restrict__ weight,
                    const float* __restrict__ weight_exp,
                    const float* __restrict__ weight_bias,
                    const float* __restrict__ running_mean,
                    const float* __restrict__ running_var,
                    const int*   __restrict__ steps_p,
                    const float* __restrict__ trace,
                    const float* __restrict__ tracesq,
                    float* __restrict__ p_rm, float* __restrict__ p_scale) {
    __shared__ float sNum[BB];
    __shared__ float sNum2[BB];
    const int t = threadIdx.x;
    if (t < BB) {
        float s = 0.f;
        for (int i = 0; i < NN; ++i) { float mv = mask[t * NN + i]; s += mv * mv; }
        sNum[t]  = s;
        sNum2[t] = fmaxf(s - 1.f, 1.f);
    }
    __syncthreads();

    float mean = 0.f, var = 0.f;
    for (int b = 0; b < BB; ++b) {
        const float tr = trace[b * FF + t];
        const float ts = tracesq[b * FF + t];
        const float n  = sNum[b];
        const float n2 = sNum2[b];
        mean += tr / n;
        var  += ts / n2 - (tr * tr) / (n * n2);
    }
    mean *= (1.f / (float)BB);
    var  *= (1.f / (float)BB);

    const int st = steps_p[0];
    float momentum;
    if (st < 100) {
        const float beta = (float)st * 0.01f;
        momentum = 0.997f * beta + 0.8f * (1.f - beta);
    } else {
        momentum = 0.997f;
    }
    const float rm = momentum * running_mean[t] + (1.f - momentum) * mean;
    const float rv = momentum * running_var[t]  + (1.f - momentum) * var;
    const float gain = weight[t] * expf(weight_exp[t]) + weight_bias[t];

    p_rm[t]    = rm;
    p_scale[t] = gain / (sqrtf(rv) + EPS);
}

// ---------------------------------------------------------------------------
// Kernel 3: streaming normalization (float4 vectorized, HBM-bound pass):
//   out[b,f,i,j] = (x - rm[f]*d_ij) * m_i * m_j * scale[f] + bias[f,i]*d_ij
// ---------------------------------------------------------------------------
__global__ void __launch_bounds__(256)
mfb_apply_kernel(const float* __restrict__ x, const float* __restrict__ mask,
                 const float* __restrict__ bias,
                 const float* __restrict__ p_rm, const float* __restrict__ p_scale,
                 float* __restrict__ out) {
    const int idx = blockIdx.x * 256 + threadIdx.x;   // float4 index
    const int j0 = (idx & 31) << 2;                   // 32 float4 per row
    const int i  = (idx >> 5) & (NN - 1);
    const int f  = (idx >> 12) & (FF - 1);
    const int b  = idx >> 18;

    const float4 xv = reinterpret_cast<const float4*>(x)[idx];
    const float4 mj = reinterpret_cast<const float4*>(mask)[(b * NN + j0) >> 2];
    const float  mi = mask[b * NN + i];
    const float  rm = p_rm[f];
    const float  sc = p_scale[f];
    const float  g  = mi * sc;
    const float  bi = bias[f * NN + i];

    float4 o;
    o.x = (xv.x - ((i == j0 + 0) ? rm : 0.f)) * (g * mj.x) + ((i == j0 + 0) ? bi : 0.f);
    o.y = (xv.y - ((i == j0 + 1) ? rm : 0.f)) * (g * mj.y) + ((i == j0 + 1) ? bi : 0.f);
    o.z = (xv.z - ((i == j0 + 2) ? rm : 0.f)) * (g * mj.z) + ((i == j0 + 2) ? bi : 0.f);
    o.w = (xv.w - ((i == j0 + 3) ? rm : 0.f)) * (g * mj.w) + ((i == j0 + 3) ? bi : 0.f);
    reinterpret_cast<float4*>(out)[idx] = o;
}

// ---------------------------------------------------------------------------
extern "C" void kernel_launch(void* const* d_in, const int* in_sizes, int n_in,
                              void* d_out, int out_size, void* d_ws, size_t ws_size,
                              hipStream_t stream) {
    const float* x           = (const float*)d_in[0];
    const float* mask        = (const float*)d_in[1];
    const float* weight      = (const float*)d_in[2];
    const float* weight_exp  = (const float*)d_in[3];
    const float* weight_bias = (const float*)d_in[4];
    const float* bias        = (const float*)d_in[5];
    const float* rmean       = (const float*)d_in[6];
    const float* rvar        = (const float*)d_in[7];
    const int*   steps       = (const int*)d_in[8];
    float* out = (float*)d_out;

    float* wsf      = (float*)d_ws;
    float* trace    = wsf;                  // [B*F]
    float* tracesq  = wsf + BB * FF;        // [B*F]
    float* p_rm     = wsf + 2 * BB * FF;    // [F]
    float* p_scale  = wsf + 2 * BB * FF + FF; // [F]

    // Pass 1: WMMA diagonal-block trace statistics
    mfb_stats_kernel<<<BB * FF, 256, 0, stream>>>(x, mask, trace, tracesq);

    // Pass 2: tiny per-feature finalize
    mfb_finalize_kernel<<<1, 64, 0, stream>>>(mask, weight, weight_exp, weight_bias,
                                              rmean, rvar, steps, trace, tracesq,
                                              p_rm, p_scale);

    // Pass 3: streaming normalize/affine (bandwidth bound)
    const int total4 = (BB * FF * NN * NN) / 4;
    mfb_apply_kernel<<<total4 / 256, 256, 0, stream>>>(x, mask, bias, p_rm, p_scale, out);
}